// GINGNN_16758962389223
// MI455X (gfx1250) — compile-verified
//
#include <hip/hip_runtime.h>

#define D      96
#define OUTD   384    // 4*D output row stride
#define NNODES 50000
#define NEDGES 800000

typedef float v2f __attribute__((ext_vector_type(2)));
typedef float v8f __attribute__((ext_vector_type(8)));

// out[n, 0:96] = x[n, :]
__global__ void gin_init_out(const float* __restrict__ x, float* __restrict__ out, int n) {
  int i = blockIdx.x * blockDim.x + threadIdx.x;
  if (i >= n * D) return;
  int r = i / D, c = i % D;
  out[(size_t)r * OUTD + c] = x[i];
}

__global__ void gin_zero(float* __restrict__ p, int n) {
  int i = blockIdx.x * blockDim.x + threadIdx.x;
  if (i < n) p[i] = 0.0f;
}

// one wave32 per edge: aggr[dst*96 + k] += h[src][k], k = lane, lane+32, lane+64
__global__ void gin_scatter(const float* __restrict__ hbase,      // out + layer*96 (row stride OUTD)
                            const long long* __restrict__ ei,     // int64 [2, E]
                            float* __restrict__ aggr, int nedges) {
  int e    = blockIdx.x * (blockDim.x >> 5) + (threadIdx.x >> 5);
  int lane = threadIdx.x & 31;
  if (e >= nedges) return;
  long long s = ei[e];
  long long d = ei[(size_t)nedges + e];
  const float* hs = hbase + (size_t)s * OUTD;
  float*       ag = aggr  + (size_t)d * D;
#pragma unroll
  for (int i = 0; i < 3; ++i)
    atomicAdd(&ag[lane + 32 * i], hs[lane + 32 * i]);
}

// Fragment-swizzled LDS layouts (all v2f reads are single conflict-free ds_load_b64):
//   Weight W (K x N):  entry ((kc*6 + nt)*32 + lane)*2 + sub
//       kc = k>>2, nt = col>>4, lane = ((k>>1)&1)*16 + (col&15), sub = k&1
//   H tile  (16 x 96): entry (kc*32 + lane)*2 + sub
//       kc = c>>2, lane = ((c>>1)&1)*16 + r, sub = c&1
__launch_bounds__(128)
__global__ void gin_mlp(const float* __restrict__ aggr,
                        const float* __restrict__ hprev,   // out + layer*96     (stride OUTD)
                        float* __restrict__ hout,          // out + (layer+1)*96 (stride OUTD)
                        const float* __restrict__ w1, const float* __restrict__ b1,
                        const float* __restrict__ w2, const float* __restrict__ b2,
                        int nrows) {
  __shared__ __align__(16) float lw1f[D * D];     // 9216 floats, fragment-ready
  __shared__ __align__(16) float lw2f[D * D];
  __shared__ float lb1[D];
  __shared__ float lb2[D];
  __shared__ __align__(16) float ltile[4][16 * D]; // per-wave A-frag-swizzled H tile

  const int tid = threadIdx.x;
  // stage weights into fragment-ready LDS layout (coalesced global reads)
  for (int i = tid; i < D * D; i += 128) {
    int k = i / D, col = i % D;
    int dst = (((k >> 2) * 6 + (col >> 4)) * 32 + ((k >> 1) & 1) * 16 + (col & 15)) * 2 + (k & 1);
    lw1f[dst] = w1[i];
    lw2f[dst] = w2[i];
  }
  if (tid < D) { lb1[tid] = b1[tid]; lb2[tid] = b2[tid]; }
  __syncthreads();

  const int wave = tid >> 5;
  const int lane = tid & 31;
  const int m0   = (blockIdx.x * 4 + wave) * 16;
  if (m0 >= nrows) return;          // wave-uniform: EXEC stays all-1s for WMMA

  float* ht = ltile[wave];
  const int l15 = lane & 15;
  const int hi  = lane >> 4;        // half-wave select (K/M split per ISA layout)

  // H tile = aggr + h_prev, written straight into A-fragment swizzle
  for (int i = lane; i < 16 * D; i += 32) {
    int r = i / D, c = i % D;
    float val = aggr[(size_t)(m0 + r) * D + c] + hprev[(size_t)(m0 + r) * OUTD + c];
    ht[((c >> 2) * 32 + ((c >> 1) & 1) * 16 + r) * 2 + (c & 1)] = val;
  }
  // wave-private tile: DS ops are in-order within a wave

  v8f acc[6];
  // ---------------- GEMM1: (16x96) @ W1(96x96) ----------------
#pragma unroll
  for (int nt = 0; nt < 6; ++nt) acc[nt] = (v8f){0,0,0,0,0,0,0,0};
  for (int kc = 0; kc < 24; ++kc) {
    v2f a = *(const v2f*)&ht[kc * 64 + lane * 2];          // one ds_load_b64
#pragma unroll
    for (int nt = 0; nt < 6; ++nt) {
      v2f b = *(const v2f*)&lw1f[((kc * 6 + nt) * 32 + lane) * 2];
      acc[nt] = __builtin_amdgcn_wmma_f32_16x16x4_f32(
          false, a, false, b, (short)0, acc[nt], false, false);
    }
  }
  // bias + ReLU, transpose back into A-fragment swizzle for GEMM2
#pragma unroll
  for (int nt = 0; nt < 6; ++nt) {
    float bv = lb1[nt * 16 + l15];
#pragma unroll
    for (int v = 0; v < 8; ++v) {
      float val = acc[nt][v] + bv;
      val = val > 0.0f ? val : 0.0f;
      int row = v + 8 * hi;                  // C layout: row = v + 8*(lane>>4)
      int col = nt * 16 + l15;               //           col = lane&15 within tile
      ht[((col >> 2) * 32 + ((col >> 1) & 1) * 16 + row) * 2 + (col & 1)] = val;
    }
  }

  // ---------------- GEMM2: (16x96) @ W2(96x96) ----------------
#pragma unroll
  for (int nt = 0; nt < 6; ++nt) acc[nt] = (v8f){0,0,0,0,0,0,0,0};
  for (int kc = 0; kc < 24; ++kc) {
    v2f a = *(const v2f*)&ht[kc * 64 + lane * 2];
#pragma unroll
    for (int nt = 0; nt < 6; ++nt) {
      v2f b = *(const v2f*)&lw2f[((kc * 6 + nt) * 32 + lane) * 2];
      acc[nt] = __builtin_amdgcn_wmma_f32_16x16x4_f32(
          false, a, false, b, (short)0, acc[nt], false, false);
    }
  }
  // bias + store straight into output slab
#pragma unroll
  for (int nt = 0; nt < 6; ++nt) {
    float bv = lb2[nt * 16 + l15];
#pragma unroll
    for (int v = 0; v < 8; ++v) {
      int row = v + 8 * hi;
      hout[(size_t)(m0 + row) * OUTD + nt * 16 + l15] = acc[nt][v] + bv;
    }
  }
}

extern "C" void kernel_launch(void* const* d_in, const int* in_sizes, int n_in,
                              void* d_out, int out_size, void* d_ws, size_t ws_size,
                              hipStream_t stream) {
  const float*     x  = (const float*)d_in[0];
  const long long* ei = (const long long*)d_in[1];
  // dict order: x, edge_index, then per layer: w1, b1, w2, b2
  const float* w1[3] = {(const float*)d_in[2],  (const float*)d_in[6],  (const float*)d_in[10]};
  const float* b1[3] = {(const float*)d_in[3],  (const float*)d_in[7],  (const float*)d_in[11]};
  const float* w2[3] = {(const float*)d_in[4],  (const float*)d_in[8],  (const float*)d_in[12]};
  const float* b2[3] = {(const float*)d_in[5],  (const float*)d_in[9],  (const float*)d_in[13]};
  float* out  = (float*)d_out;
  float* aggr = (float*)d_ws;                   // N*D floats = 19.2 MB scratch

  (void)in_sizes; (void)n_in; (void)out_size; (void)ws_size;

  gin_init_out<<<(NNODES * D + 255) / 256, 256, 0, stream>>>(x, out, NNODES);
  for (int l = 0; l < 3; ++l) {
    gin_zero<<<(NNODES * D + 255) / 256, 256, 0, stream>>>(aggr, NNODES * D);
    gin_scatter<<<(NEDGES + 7) / 8, 256, 0, stream>>>(out + l * D, ei, aggr, NEDGES);
    const int tiles = NNODES / 16;              // 3125 row tiles
    gin_mlp<<<(tiles + 3) / 4, 128, 0, stream>>>(aggr, out + l * D, out + (l + 1) * D,
                                                 w1[l], b1[l], w2[l], b2[l], NNODES);
  }
}